// GNN_17008070492329
// MI455X (gfx1250) — compile-verified
//
#include <hip/hip_runtime.h>

typedef float v2f __attribute__((ext_vector_type(2)));
typedef float v8f __attribute__((ext_vector_type(8)));

// ---------------- utility ----------------
__global__ void fill_f32(float* __restrict__ p, float v, int n) {
    int i = blockIdx.x * blockDim.x + threadIdx.x;
    if (i < n) p[i] = v;
}

// deg[col[e]] += 1 (deg pre-initialized to 1.0 for the self loop)
__global__ void deg_count(const int* __restrict__ col, float* __restrict__ deg, int E) {
    int e = blockIdx.x * blockDim.x + threadIdx.x;
    if (e < E) atomicAdd(&deg[col[e]], 1.0f);
}

__global__ void deg_rsqrt(float* __restrict__ deg, int n) {
    int i = blockIdx.x * blockDim.x + threadIdx.x;
    if (i < n) deg[i] = rsqrtf(deg[i]);   // deg >= 1 always (self loop)
}

// ---------------- WMMA node-feature transforms ----------------
// Layer 1: [16 nodes x 3] @ W1[3,16] -> [16 x 16], K padded 3->4, one
// V_WMMA_F32_16X16X4_F32 per 16-node tile (one wave32 per tile).
// A layout (16x4 f32): lane half h: v0 -> K=2h, v1 -> K=2h+1, row M = lane%16.
// B layout (4x16 f32): lane half h: v0 -> K=2h, v1 -> K=2h+1, col N = lane%16.
// C/D (16x16 f32): VGPR i: row M = i + 8h, col N = lane%16.
__global__ void lin1_wmma(const float* __restrict__ x, const float* __restrict__ W1,
                          float* __restrict__ out, int n_nodes) {
    int wave = (blockIdx.x * blockDim.x + threadIdx.x) >> 5;
    int lane = threadIdx.x & 31;
    int base = wave << 4;
    if (base >= n_nodes) return;               // whole-wave exit: EXEC stays all-1s
    int r  = lane & 15;
    int hh = lane >> 4;
    const float* xr = x + (size_t)(base + r) * 3;
    v2f a, b;
    a.x = xr[2 * hh];                          // K = 0 or 2 (both < 3)
    a.y = hh ? 0.0f : xr[1];                   // K = 1, K = 3 is zero pad
    b.x = W1[(2 * hh) * 16 + r];
    b.y = hh ? 0.0f : W1[16 + r];
    v8f c = {};
    c = __builtin_amdgcn_wmma_f32_16x16x4_f32(false, a, false, b, (short)0, c, false, false);
    float* o = out + (size_t)base * 16;
#pragma unroll
    for (int i = 0; i < 8; ++i) o[(i + 8 * hh) * 16 + r] = c[i];
}

// Layer 2: [16 nodes x 16] @ W[16,16] -> [16 x 16]; K=16 as 4 chained WMMAs.
__global__ void lin16_wmma(const float* __restrict__ in, const float* __restrict__ W,
                           float* __restrict__ out, int n_nodes) {
    int wave = (blockIdx.x * blockDim.x + threadIdx.x) >> 5;
    int lane = threadIdx.x & 31;
    int base = wave << 4;
    if (base >= n_nodes) return;
    int r  = lane & 15;
    int hh = lane >> 4;
    const float* ar = in + (size_t)(base + r) * 16;
    v8f c = {};
#pragma unroll
    for (int kb = 0; kb < 16; kb += 4) {
        v2f a, b;
        a.x = ar[kb + 2 * hh];
        a.y = ar[kb + 2 * hh + 1];
        b.x = W[(kb + 2 * hh) * 16 + r];
        b.y = W[(kb + 2 * hh + 1) * 16 + r];
        c = __builtin_amdgcn_wmma_f32_16x16x4_f32(false, a, false, b, (short)0, c, false, false);
    }
    float* o = out + (size_t)base * 16;
#pragma unroll
    for (int i = 0; i < 8; ++i) o[(i + 8 * hh) * 16 + r] = c[i];
}

// ---------------- edge propagation (bandwidth/atomic bound) ----------------
__global__ void edge_scatter(const int* __restrict__ row, const int* __restrict__ col,
                             const float* __restrict__ dinv, const float* __restrict__ h,
                             float* __restrict__ out, int E) {
    int e = blockIdx.x * blockDim.x + threadIdx.x;
    if (e >= E) return;
    int s = row[e], t = col[e];
    float w = dinv[s] * dinv[t];
    const float4* hs = (const float4*)(h + (size_t)s * 16);   // 64B-aligned gather
    float* ot = out + (size_t)t * 16;                          // hot 6.4MB set lives in L2
#pragma unroll
    for (int q = 0; q < 4; ++q) {
        float4 v = hs[q];
        atomicAdd(ot + 4 * q + 0, v.x * w);
        atomicAdd(ot + 4 * q + 1, v.y * w);
        atomicAdd(ot + 4 * q + 2, v.z * w);
        atomicAdd(ot + 4 * q + 3, v.w * w);
    }
}

// out += h * dinv^2 (self loop) + bias, optional ReLU
__global__ void node_finalize(const float* __restrict__ h, const float* __restrict__ dinv,
                              const float* __restrict__ bias, float* __restrict__ out,
                              int n_nodes, int relu) {
    int idx = blockIdx.x * blockDim.x + threadIdx.x;
    if (idx >= n_nodes * 16) return;
    int n = idx >> 4, f = idx & 15;
    float di = dinv[n];
    float v = out[idx] + h[idx] * di * di + bias[f];
    out[idx] = relu ? fmaxf(v, 0.0f) : v;
}

// global_add_pool: nodes -> graphs
__global__ void pool_kernel(const float* __restrict__ a, const int* __restrict__ batch,
                            float* __restrict__ g, int n_nodes) {
    int idx = blockIdx.x * blockDim.x + threadIdx.x;
    if (idx >= n_nodes * 16) return;
    int n = idx >> 4, f = idx & 15;
    atomicAdd(&g[(size_t)batch[n] * 16 + f], a[idx]);
}

// logits = g @ Wlin + blin; log_softmax over 7 classes; one thread per graph
__global__ void head_kernel(const float* __restrict__ g, const float* __restrict__ Wlin,
                            const float* __restrict__ blin, float* __restrict__ out, int G) {
    int i = blockIdx.x * blockDim.x + threadIdx.x;
    if (i >= G) return;
    float feat[16];
#pragma unroll
    for (int f = 0; f < 16; ++f) feat[f] = g[(size_t)i * 16 + f];
    float logit[7];
#pragma unroll
    for (int c = 0; c < 7; ++c) {
        float acc = blin[c];
#pragma unroll
        for (int f = 0; f < 16; ++f) acc += feat[f] * Wlin[f * 7 + c];
        logit[c] = acc;
    }
    float m = logit[0];
#pragma unroll
    for (int c = 1; c < 7; ++c) m = fmaxf(m, logit[c]);
    float s = 0.0f;
#pragma unroll
    for (int c = 0; c < 7; ++c) s += expf(logit[c] - m);
    float lse = m + logf(s);
#pragma unroll
    for (int c = 0; c < 7; ++c) out[(size_t)i * 7 + c] = logit[c] - lse;
}

extern "C" void kernel_launch(void* const* d_in, const int* in_sizes, int n_in,
                              void* d_out, int out_size, void* d_ws, size_t ws_size,
                              hipStream_t stream) {
    (void)n_in; (void)ws_size;
    const float* x     = (const float*)d_in[0];
    const int*   ei    = (const int*)d_in[1];
    const int*   batch = (const int*)d_in[2];
    const float* W1    = (const float*)d_in[4];
    const float* b1    = (const float*)d_in[5];
    const float* W2    = (const float*)d_in[6];
    const float* b2    = (const float*)d_in[7];
    const float* Wlin  = (const float*)d_in[8];
    const float* blin  = (const float*)d_in[9];
    float* out = (float*)d_out;

    const int N = in_sizes[0] / 3;     // 100000
    const int E = in_sizes[1] / 2;     // 3200000
    const int G = out_size / 7;        // 1024

    const int* row = ei;               // sources
    const int* col = ei + E;           // targets

    // workspace carve-out (256B aligned): dinv[N] | bufA[N*16] | bufB[N*16] | g[G*16]
    char* ws = (char*)d_ws;
    size_t off = 0;
    auto carve = [&](size_t bytes) -> float* {
        float* p = (float*)(ws + off);
        off += (bytes + 255) & ~(size_t)255;
        return p;
    };
    float* dinv  = carve((size_t)N * 4);
    float* bufA  = carve((size_t)N * 16 * 4);
    float* bufB  = carve((size_t)N * 16 * 4);
    float* gpool = carve((size_t)G * 16 * 4);

    const int B = 256;                         // 8 wave32s per block
    const int tiles = (N + 15) / 16;           // one wave per 16-node tile
    const int gemm_blocks = (tiles + (B / 32) - 1) / (B / 32);

    // symmetric normalization (shared by both layers; self loop folded into init=1)
    fill_f32 <<<(N + B - 1) / B, B, 0, stream>>>(dinv, 1.0f, N);
    deg_count<<<(E + B - 1) / B, B, 0, stream>>>(col, dinv, E);
    deg_rsqrt<<<(N + B - 1) / B, B, 0, stream>>>(dinv, N);

    // ---- layer 1 ----
    lin1_wmma   <<<gemm_blocks, B, 0, stream>>>(x, W1, bufA, N);
    fill_f32    <<<(N * 16 + B - 1) / B, B, 0, stream>>>(bufB, 0.0f, N * 16);
    edge_scatter<<<(E + B - 1) / B, B, 0, stream>>>(row, col, dinv, bufA, bufB, E);
    node_finalize<<<(N * 16 + B - 1) / B, B, 0, stream>>>(bufA, dinv, b1, bufB, N, 1);

    // ---- layer 2 ----
    lin16_wmma  <<<gemm_blocks, B, 0, stream>>>(bufB, W2, bufA, N);
    fill_f32    <<<(N * 16 + B - 1) / B, B, 0, stream>>>(bufB, 0.0f, N * 16);
    edge_scatter<<<(E + B - 1) / B, B, 0, stream>>>(row, col, dinv, bufA, bufB, E);
    node_finalize<<<(N * 16 + B - 1) / B, B, 0, stream>>>(bufA, dinv, b2, bufB, N, 0);

    // ---- pool + head ----
    fill_f32   <<<(G * 16 + B - 1) / B, B, 0, stream>>>(gpool, 0.0f, G * 16);
    pool_kernel<<<(N * 16 + B - 1) / B, B, 0, stream>>>(bufB, batch, gpool, N);
    head_kernel<<<(G + B - 1) / B, B, 0, stream>>>(gpool, Wlin, blin, out, G);
}